// GATNet_34488587387426
// MI455X (gfx1250) — compile-verified
//
#include <hip/hip_runtime.h>
#include <hip/hip_bf16.h>

// ---------------- problem constants (match reference) ----------------
static constexpr int NN   = 20000;   // nodes
static constexpr int EE   = 400000;  // directed edges (before self loops)
static constexpr int ET   = EE + NN; // edges incl. self loops = 420000
static constexpr int GG   = 64;      // graphs
static constexpr int VIN  = 128;
static constexpr int CONV = 128;
static constexpr int H1   = 4;       // gat1 heads
static constexpr int HC   = H1 * CONV; // 512
static constexpr int NET  = 256;
static constexpr int NX   = 9;

typedef float v2f __attribute__((ext_vector_type(2)));
typedef float v8f __attribute__((ext_vector_type(8)));

// ---------------- utility fills ----------------
__global__ void fill_f32(float* p, float v, size_t n) {
    size_t i = (size_t)blockIdx.x * blockDim.x + threadIdx.x;
    if (i < n) p[i] = v;
}
__global__ void fill_u32(unsigned* p, unsigned v, size_t n) {
    size_t i = (size_t)blockIdx.x * blockDim.x + threadIdx.x;
    if (i < n) p[i] = v;
}

// ---------------- fp32 WMMA GEMM: D = A[M,K] @ B[K,N] ----------------
// one wave per 16x16 tile; V_WMMA_F32_16X16X4_F32, K stepped by 4.
__global__ __launch_bounds__(256) void wmma_gemm_f32(const float* __restrict__ A,
                                                     const float* __restrict__ B,
                                                     float* __restrict__ D,
                                                     int M, int Nn, int K) {
    int gid   = blockIdx.x * blockDim.x + threadIdx.x;
    int wave  = gid >> 5;
    int lane  = threadIdx.x & 31;
    int tilesN = Nn >> 4;
    int tm = wave / tilesN;
    int tn = wave % tilesN;
    if (tm * 16 >= M) return;            // whole wave exits together (EXEC stays all-1)

    int hi = lane >> 4;                   // 0: K pair {0,1}, 1: K pair {2,3}
    int l  = lane & 15;                   // M (for A) / N (for B,C,D)

    v8f acc = {};
    const float* Arow = A + (size_t)(tm * 16 + l) * K;
    const float* Bbase = B + tn * 16 + l;
    for (int k0 = 0; k0 < K; k0 += 4) {
        v2f a, b;
        a[0] = Arow[k0 + hi * 2 + 0];
        a[1] = Arow[k0 + hi * 2 + 1];
        const float* Bp = Bbase + (size_t)(k0 + hi * 2) * Nn;
        b[0] = Bp[0];
        b[1] = Bp[Nn];
        acc = __builtin_amdgcn_wmma_f32_16x16x4_f32(false, a, false, b,
                                                    (short)0, acc, false, false);
    }
    float* Dt = D + (size_t)(tm * 16) * Nn + tn * 16;
#pragma unroll
    for (int r = 0; r < 8; ++r)
        Dt[(size_t)(r + hi * 8) * Nn + l] = acc[r];
}

// ---------------- attention logits: as[n,h]=<h_row, a_src[h]>, ad likewise ----
// one wave per (node, head); lanes stride channels, shuffle-reduce (wave32).
__global__ __launch_bounds__(256) void alpha_kernel(const float* __restrict__ h,
                                                    const float* __restrict__ a_src,
                                                    const float* __restrict__ a_dst,
                                                    float* __restrict__ as,
                                                    float* __restrict__ ad,
                                                    int H, int C) {
    int gid  = blockIdx.x * blockDim.x + threadIdx.x;
    int wave = gid >> 5;
    int lane = threadIdx.x & 31;
    int n  = wave / H;
    int hd = wave % H;
    if (n >= NN) return;
    const float* row = h + (size_t)n * H * C + (size_t)hd * C;
    float s = 0.f, d = 0.f;
    for (int c = lane; c < C; c += 32) {
        float v = row[c];
        s += v * a_src[hd * C + c];
        d += v * a_dst[hd * C + c];
    }
#pragma unroll
    for (int o = 16; o > 0; o >>= 1) {
        s += __shfl_xor(s, o, 32);
        d += __shfl_xor(d, o, 32);
    }
    if (lane == 0) { as[n * H + hd] = s; ad[n * H + hd] = d; }
}

__device__ __forceinline__ int esrc(const int* ei, int e) { return (e < EE) ? ei[e] : (e - EE); }
__device__ __forceinline__ int edst(const int* ei, int e) { return (e < EE) ? ei[EE + e] : (e - EE); }

__device__ __forceinline__ float lrelu(float x) { return x > 0.f ? x : 0.2f * x; }
__device__ __forceinline__ unsigned f2key(float f) {
    unsigned u = __float_as_uint(f);
    return (u & 0x80000000u) ? ~u : (u | 0x80000000u);
}
__device__ __forceinline__ float key2f(unsigned k) {
    return (k & 0x80000000u) ? __uint_as_float(k & 0x7FFFFFFFu) : __uint_as_float(~k);
}

// ---------------- edge pass 1: segment max of logits per dst ----------------
__global__ __launch_bounds__(256) void edge_max(const int* __restrict__ ei,
                                                const float* __restrict__ as,
                                                const float* __restrict__ ad,
                                                unsigned* __restrict__ mkey, int H) {
    int t = blockIdx.x * blockDim.x + threadIdx.x;
    if (t >= ET * H) return;
    int e = t / H, hd = t % H;
    int s = esrc(ei, e), d = edst(ei, e);
    float x = lrelu(as[s * H + hd] + ad[d * H + hd]);
    atomicMax(&mkey[d * H + hd], f2key(x));
}

// ---------------- edge pass 2: p = exp(logit - max); denom += p ----------------
__global__ __launch_bounds__(256) void edge_exp(const int* __restrict__ ei,
                                                const float* __restrict__ as,
                                                const float* __restrict__ ad,
                                                const unsigned* __restrict__ mkey,
                                                float* __restrict__ p,
                                                float* __restrict__ denom, int H) {
    int t = blockIdx.x * blockDim.x + threadIdx.x;
    if (t >= ET * H) return;
    int e = t / H, hd = t % H;
    int s = esrc(ei, e), d = edst(ei, e);
    float x  = lrelu(as[s * H + hd] + ad[d * H + hd]);
    float pe = expf(x - key2f(mkey[d * H + hd]));
    p[(size_t)e * H + hd] = pe;
    atomicAdd(&denom[d * H + hd], pe);
}

// ---------------- edge pass 3: out[dst,h,:] += alpha * h[src,h,:] ----------------
// one wave per (edge, head); lanes stride the C channels (coalesced atomics).
__global__ __launch_bounds__(256) void edge_scatter(const int* __restrict__ ei,
                                                    const float* __restrict__ p,
                                                    const float* __restrict__ denom,
                                                    const float* __restrict__ h,
                                                    float* __restrict__ out,
                                                    int H, int C) {
    int gid  = blockIdx.x * blockDim.x + threadIdx.x;
    int wave = gid >> 5;
    int lane = threadIdx.x & 31;
    int e  = wave / H;
    int hd = wave % H;
    if (e >= ET) return;
    int s = esrc(ei, e), d = edst(ei, e);
    float alpha = p[(size_t)e * H + hd] / (denom[d * H + hd] + 1e-16f);
    const float* hrow = h   + (size_t)s * H * C + (size_t)hd * C;
    float*       orow = out + (size_t)d * H * C + (size_t)hd * C;
    for (int c = lane; c < C; c += 32)
        atomicAdd(&orow[c], alpha * hrow[c]);
}

// ---------------- batchnorm stats (per channel over N rows) ----------------
__global__ __launch_bounds__(256) void bn_stats(const float* __restrict__ x,
                                                float* __restrict__ mu,
                                                float* __restrict__ rsig, int C) {
    int c = blockIdx.x;
    float s = 0.f, s2 = 0.f;
    for (int n = threadIdx.x; n < NN; n += blockDim.x) {
        float v = x[(size_t)n * C + c];
        s += v; s2 += v * v;
    }
    __shared__ float sh[256], sh2[256];
    int t = threadIdx.x;
    sh[t] = s; sh2[t] = s2;
    __syncthreads();
    for (int o = 128; o > 0; o >>= 1) {
        if (t < o) { sh[t] += sh[t + o]; sh2[t] += sh2[t + o]; }
        __syncthreads();
    }
    if (t == 0) {
        float m   = sh[0] / (float)NN;
        float var = sh2[0] / (float)NN - m * m;
        mu[c]   = m;
        rsig[c] = rsqrtf(var + 1e-5f);
    }
}

// ---------------- batchnorm apply (in place) + ReLU ----------------
__global__ __launch_bounds__(256) void bn_apply_relu(float* __restrict__ x,
                                                     const float* __restrict__ mu,
                                                     const float* __restrict__ rsig,
                                                     const float* __restrict__ gamma,
                                                     const float* __restrict__ beta, int C) {
    size_t i = (size_t)blockIdx.x * blockDim.x + threadIdx.x;
    if (i >= (size_t)NN * C) return;
    int c = (int)(i % C);
    float v = gamma[c] * (x[i] - mu[c]) * rsig[c] + beta[c];
    x[i] = v > 0.f ? v : 0.f;
}

// ---------------- global add pool ----------------
__global__ __launch_bounds__(256) void pool_kernel(const float* __restrict__ h,
                                                   const int* __restrict__ batch,
                                                   float* __restrict__ g) {
    size_t i = (size_t)blockIdx.x * blockDim.x + threadIdx.x;
    if (i >= (size_t)NN * CONV) return;
    int n = (int)(i / CONV), c = (int)(i % CONV);
    atomicAdd(&g[batch[n] * CONV + c], h[i]);
}

// ---------------- tiny MLP head: one block per graph ----------------
__global__ __launch_bounds__(256) void mlp_kernel(const float* __restrict__ g,
                                                  const float* __restrict__ extras,
                                                  const float* __restrict__ Wm1, const float* __restrict__ bm1,
                                                  const float* __restrict__ Wm2, const float* __restrict__ bm2,
                                                  const float* __restrict__ Wm3, const float* __restrict__ bm3,
                                                  float* __restrict__ out) {
    constexpr int D0 = CONV + NX;  // 137
    constexpr int D1 = NET + NX;   // 265
    __shared__ float z0[D0];
    __shared__ float z1[D1];
    __shared__ float z2[NET];
    __shared__ float red[256];
    int gr = blockIdx.x;
    int t  = threadIdx.x;
    for (int i = t; i < D0; i += blockDim.x)
        z0[i] = (i < CONV) ? g[gr * CONV + i] : extras[gr * NX + (i - CONV)];
    __syncthreads();
    for (int j = t; j < D1; j += blockDim.x) {
        float a = bm1[j];
        for (int i = 0; i < D0; ++i) a += z0[i] * Wm1[i * D1 + j];
        z1[j] = fmaxf(a, 0.f);
    }
    __syncthreads();
    for (int j = t; j < NET; j += blockDim.x) {
        float a = bm2[j];
        for (int i = 0; i < D1; ++i) a += z1[i] * Wm2[i * NET + j];
        z2[j] = fmaxf(a, 0.f);
    }
    __syncthreads();
    float a = 0.f;
    for (int j = t; j < NET; j += blockDim.x) a += z2[j] * Wm3[j];
    red[t] = a;
    __syncthreads();
    for (int o = 128; o > 0; o >>= 1) {
        if (t < o) red[t] += red[t + o];
        __syncthreads();
    }
    if (t == 0) out[gr] = red[0] + bm3[0];
}

// ---------------- launcher ----------------
extern "C" void kernel_launch(void* const* d_in, const int* in_sizes, int n_in,
                              void* d_out, int out_size, void* d_ws, size_t ws_size,
                              hipStream_t stream) {
    const float* x      = (const float*)d_in[0];
    const int*   ei     = (const int*)  d_in[1];
    const int*   batch  = (const int*)  d_in[2];
    const float* extras = (const float*)d_in[3];
    const float* W1     = (const float*)d_in[4];
    const float* a_src1 = (const float*)d_in[5];
    const float* a_dst1 = (const float*)d_in[6];
    // d_in[7] = b1 (zeros; cancels exactly through BN)
    const float* W2     = (const float*)d_in[8];
    const float* a_src2 = (const float*)d_in[9];
    const float* a_dst2 = (const float*)d_in[10];
    // d_in[11] = b2 (zeros; cancels exactly through BN)
    const float* gamma1 = (const float*)d_in[12];
    const float* beta1  = (const float*)d_in[13];
    const float* gamma2 = (const float*)d_in[14];
    const float* beta2  = (const float*)d_in[15];
    const float* Wm1 = (const float*)d_in[16];
    const float* bm1 = (const float*)d_in[17];
    const float* Wm2 = (const float*)d_in[18];
    const float* bm2 = (const float*)d_in[19];
    const float* Wm3 = (const float*)d_in[20];
    const float* bm3 = (const float*)d_in[21];
    float* out = (float*)d_out;

    // workspace layout (floats)
    float* ws = (float*)d_ws;
    size_t o = 0;
    float*    h1   = ws + o; o += (size_t)NN * HC;   // h1 [N,512]; later h2pre [N,128] + out2 [N,128]
    float*    out1 = ws + o; o += (size_t)NN * HC;   // out1 [N,512]; bn1 in place
    float*    as1  = ws + o; o += (size_t)NN * H1;
    float*    ad1  = ws + o; o += (size_t)NN * H1;
    unsigned* m1   = (unsigned*)(ws + o); o += (size_t)NN * H1;
    float*    den1 = ws + o; o += (size_t)NN * H1;
    float*    p1   = ws + o; o += (size_t)ET * H1;
    float*    as2  = ws + o; o += NN;
    float*    ad2  = ws + o; o += NN;
    unsigned* m2   = (unsigned*)(ws + o); o += NN;
    float*    den2 = ws + o; o += NN;
    float*    p2   = ws + o; o += ET;
    float*    mu1  = ws + o; o += HC;
    float*    rs1  = ws + o; o += HC;
    float*    mu2  = ws + o; o += CONV;
    float*    rs2  = ws + o; o += CONV;
    float*    gp   = ws + o; o += (size_t)GG * CONV;

    float* h2pre = h1;                       // [N,128] reuses h1 (dead after scatter1)
    float* out2  = h1 + (size_t)NN * CONV;   // [N,128]

    const int TB = 256;
    auto cdiv = [](size_t a, size_t b) { return (unsigned)((a + b - 1) / b); };

    // ---- zero accumulators (graph-capture safe fills) ----
    fill_f32<<<cdiv((size_t)NN * HC, TB), TB, 0, stream>>>(out1, 0.f, (size_t)NN * HC);
    fill_u32<<<cdiv((size_t)NN * H1, TB), TB, 0, stream>>>(m1, 0u, (size_t)NN * H1);
    fill_f32<<<cdiv((size_t)NN * H1, TB), TB, 0, stream>>>(den1, 0.f, (size_t)NN * H1);
    fill_u32<<<cdiv((size_t)NN, TB), TB, 0, stream>>>(m2, 0u, (size_t)NN);
    fill_f32<<<cdiv((size_t)NN, TB), TB, 0, stream>>>(den2, 0.f, (size_t)NN);
    fill_f32<<<cdiv((size_t)GG * CONV, TB), TB, 0, stream>>>(gp, 0.f, (size_t)GG * CONV);

    // ---- GAT layer 1 ----
    {   // GEMM1: h1 = x @ W1   [20000,128]x[128,512]
        unsigned waves = (NN / 16) * (HC / 16);              // 40000
        wmma_gemm_f32<<<cdiv((size_t)waves * 32, TB), TB, 0, stream>>>(x, W1, h1, NN, HC, VIN);
    }
    {   unsigned waves = NN * H1;                            // 80000
        alpha_kernel<<<cdiv((size_t)waves * 32, TB), TB, 0, stream>>>(h1, a_src1, a_dst1, as1, ad1, H1, CONV);
    }
    edge_max<<<cdiv((size_t)ET * H1, TB), TB, 0, stream>>>(ei, as1, ad1, m1, H1);
    edge_exp<<<cdiv((size_t)ET * H1, TB), TB, 0, stream>>>(ei, as1, ad1, m1, p1, den1, H1);
    edge_scatter<<<cdiv((size_t)ET * H1 * 32, TB), TB, 0, stream>>>(ei, p1, den1, h1, out1, H1, CONV);
    bn_stats<<<HC, TB, 0, stream>>>(out1, mu1, rs1, HC);
    bn_apply_relu<<<cdiv((size_t)NN * HC, TB), TB, 0, stream>>>(out1, mu1, rs1, gamma1, beta1, HC);

    // ---- GAT layer 2 ----
    {   // GEMM2: h2pre = out1 @ W2   [20000,512]x[512,128]
        unsigned waves = (NN / 16) * (CONV / 16);            // 10000
        wmma_gemm_f32<<<cdiv((size_t)waves * 32, TB), TB, 0, stream>>>(out1, W2, h2pre, NN, CONV, HC);
    }
    {   unsigned waves = NN;                                 // H=1
        alpha_kernel<<<cdiv((size_t)waves * 32, TB), TB, 0, stream>>>(h2pre, a_src2, a_dst2, as2, ad2, 1, CONV);
    }
    fill_f32<<<cdiv((size_t)NN * CONV, TB), TB, 0, stream>>>(out2, 0.f, (size_t)NN * CONV);
    edge_max<<<cdiv((size_t)ET, TB), TB, 0, stream>>>(ei, as2, ad2, m2, 1);
    edge_exp<<<cdiv((size_t)ET, TB), TB, 0, stream>>>(ei, as2, ad2, m2, p2, den2, 1);
    edge_scatter<<<cdiv((size_t)ET * 32, TB), TB, 0, stream>>>(ei, p2, den2, h2pre, out2, 1, CONV);
    bn_stats<<<CONV, TB, 0, stream>>>(out2, mu2, rs2, CONV);
    bn_apply_relu<<<cdiv((size_t)NN * CONV, TB), TB, 0, stream>>>(out2, mu2, rs2, gamma2, beta2, CONV);

    // ---- pool + MLP head ----
    pool_kernel<<<cdiv((size_t)NN * CONV, TB), TB, 0, stream>>>(out2, batch, gp);
    mlp_kernel<<<GG, TB, 0, stream>>>(gp, extras, Wm1, bm1, Wm2, bm2, Wm3, bm3, out);
}